// ParametricConvLayer_57303453663818
// MI455X (gfx1250) — compile-verified
//
#include <hip/hip_runtime.h>

// Complex 'full' FIR (chromatic-dispersion filter) as a Toeplitz GEMM on the
// CDNA5 fp32 matrix pipe: V_WMMA_F32_16X16X4_F32, with async global->LDS
// staging (ASYNCcnt path) for interior blocks.
//
//   y_r(n) = sum_k x_r(n+k-M) w_r(k) - x_i(n+k-M) w_i(k)
//   y_i(n) = sum_k x_i(n+k-M) w_r(k) + x_r(n+k-M) w_i(k)
//
// Per 256-output chunk:  C[u,v] = y(b + 16u + v)
//   A[u,kk] = x(b + 16u + kk - M)        (16 x 272 sliding window, M = K-1)
//   B[kk,v] = w(kk - v) banded Toeplitz  (272 x 16), kk-dim = K + 15 = 272
// => 68 WMMA(K=4) steps per GEMM chain, 4 chains per complex chunk (~6% over
//    the minimal MAC count).

typedef float v2f __attribute__((ext_vector_type(2)));
typedef float v8f __attribute__((ext_vector_type(8)));

#define CHUNK      256                 // outputs per wave = one 16x16 C tile
#define WAVES      8                   // waves per block
#define GROUP_OUT  (CHUNK * WAVES)     // 2048 outputs per block
#define XTILE      (GROUP_OUT + 256)   // staged x window per channel (floats)
#define WPAD       320                 // zero-padded taps: [16, 16+K) valid

#if __has_builtin(__builtin_amdgcn_global_load_async_to_lds_b128)
#define HAVE_ASYNC_LDS 1
// Builtin expects: (v4i addrspace(1)*, v4i addrspace(3)*, imm offset, imm cpol)
typedef int v4i __attribute__((vector_size(4 * sizeof(int))));
typedef v4i __attribute__((address_space(1)))* gv4p;
typedef v4i __attribute__((address_space(3)))* lv4p;
#else
#define HAVE_ASYNC_LDS 0
#endif

static __device__ __forceinline__ void wait_async_zero() {
#if __has_builtin(__builtin_amdgcn_s_wait_asynccnt)
    __builtin_amdgcn_s_wait_asynccnt(0);
#else
    asm volatile("s_wait_asynccnt 0" ::: "memory");
#endif
}

__global__ __launch_bounds__(256) void cd_fir_wmma_f32(
    const float* __restrict__ xr, const float* __restrict__ xi,
    const float* __restrict__ wr, const float* __restrict__ wi,
    float* __restrict__ out,
    int n_len, int k_len, int out_len, int total_chunks)
{
    __shared__ float s_xr[XTILE];
    __shared__ float s_xi[XTILE];
    __shared__ float s_wr[WPAD];
    __shared__ float s_wi[WPAD];

    const int tid        = threadIdx.x;
    const int group_base = blockIdx.x * GROUP_OUT;   // first output of block
    const int x0         = group_base - 256;         // x index behind s_x[0]

    // ---- stage signal window into LDS ----
    const bool interior = (x0 >= 0) && (x0 + XTILE <= n_len);
#if HAVE_ASYNC_LDS
    if (interior) {
        // Direct async DMA global->LDS, 16B per lane per op (ASYNCcnt-tracked).
        // x0 is a multiple of 256 floats -> 16B aligned on both sides.
        for (int q4 = tid; q4 < XTILE / 4; q4 += 256) {
            const int fo = 4 * q4;
            __builtin_amdgcn_global_load_async_to_lds_b128(
                (gv4p)(xr + x0 + fo), (lv4p)&s_xr[fo], 0, 0);
            __builtin_amdgcn_global_load_async_to_lds_b128(
                (gv4p)(xi + x0 + fo), (lv4p)&s_xi[fo], 0, 0);
        }
        wait_async_zero();
    } else
#endif
    {
        // Edge blocks (and fallback): bounds-checked, zero-filled.
        for (int idx = tid; idx < XTILE; idx += 256) {
            int  g  = x0 + idx;
            bool ok = (g >= 0) && (g < n_len);
            s_xr[idx] = ok ? xr[g] : 0.0f;
            s_xi[idx] = ok ? xi[g] : 0.0f;
        }
    }
    // ---- stage taps with 16-zero head / tail padding (branch-free B gather) ----
    for (int t = tid; t < WPAD; t += 256) {
        int  k  = t - 16;
        bool ok = (k >= 0) && (k < k_len);
        s_wr[t] = ok ? wr[k] : 0.0f;
        s_wi[t] = ok ? wi[k] : 0.0f;
    }
    __syncthreads();

    const int wv    = tid >> 5;
    const int chunk = blockIdx.x * WAVES + wv;
    if (chunk >= total_chunks) return;               // wave-uniform: EXEC stays all-1s

    const int lane = tid & 31;
    const int half = lane >> 4;                      // 0: lanes 0-15, 1: lanes 16-31
    const int col  = lane & 15;

    // A 16x4 fp32 layout: lanes0-15 hold K={0,1} in VGPR{0,1}; lanes16-31 K={2,3}.
    // 8-byte aligned float2 load: s_x[256*wv + 16*m + 4q + 2*half + {0,1}]
    const int a_base  = 256 * wv + 16 * col + 2 * half;
    // B 4x16 fp32 layout: VGPR0 rows K={0,1} across lane halves, VGPR1 rows K={2,3}.
    // tap = kk - v = 4q + krow - col,  krow in {half, half+2};  +16 = pad offset.
    const int b_base  = 16 + half - col;

    v8f acc_rp = {};   // A_r * B_wr
    v8f acc_rm = {};   // A_i * B_wi   (subtracted at the end: f32 WMMA has no A/B neg)
    v8f acc_i  = {};   // A_i * B_wr + A_r * B_wi

    const int nq = (k_len + 15 + 3) >> 2;            // 68 for K = 257

    #pragma unroll 4
    for (int q = 0; q < nq; ++q) {
        const int kk = 4 * q;
        v2f a_r = *(const v2f*)&s_xr[a_base + kk];
        v2f a_i = *(const v2f*)&s_xi[a_base + kk];
        v2f b_r, b_i;
        b_r.x = s_wr[b_base + kk];
        b_r.y = s_wr[b_base + kk + 2];
        b_i.x = s_wi[b_base + kk];
        b_i.y = s_wi[b_base + kk + 2];

        acc_rp = __builtin_amdgcn_wmma_f32_16x16x4_f32(false, a_r, false, b_r,
                                                       (short)0, acc_rp, false, false);
        acc_rm = __builtin_amdgcn_wmma_f32_16x16x4_f32(false, a_i, false, b_i,
                                                       (short)0, acc_rm, false, false);
        acc_i  = __builtin_amdgcn_wmma_f32_16x16x4_f32(false, a_i, false, b_r,
                                                       (short)0, acc_i,  false, false);
        acc_i  = __builtin_amdgcn_wmma_f32_16x16x4_f32(false, a_r, false, b_i,
                                                       (short)0, acc_i,  false, false);
    }

    // C 16x16 f32 layout: VGPR j -> row M = j + 8*half, column N = col.
    const int chunk_base = chunk * CHUNK;
    #pragma unroll
    for (int j = 0; j < 8; ++j) {
        int n = chunk_base + 16 * (j + 8 * half) + col;
        if (n < out_len) {
            out[n]           = acc_rp[j] - acc_rm[j];   // real channel (row 0)
            out[out_len + n] = acc_i[j];                // imag channel (row 1)
        }
    }
}

extern "C" void kernel_launch(void* const* d_in, const int* in_sizes, int n_in,
                              void* d_out, int out_size, void* d_ws, size_t ws_size,
                              hipStream_t stream) {
    const float* xr = (const float*)d_in[0];
    const float* xi = (const float*)d_in[1];
    const float* wr = (const float*)d_in[2];
    const float* wi = (const float*)d_in[3];
    float*       out = (float*)d_out;

    const int n_len   = in_sizes[0];
    const int k_len   = in_sizes[2];
    const int out_len = n_len + k_len - 1;                  // 8,388,864
    const int total_chunks = (out_len + CHUNK - 1) / CHUNK; // 32,769
    const int grid         = (total_chunks + WAVES - 1) / WAVES;

    cd_fir_wmma_f32<<<grid, 256, 0, stream>>>(xr, xi, wr, wi, out,
                                              n_len, k_len, out_len, total_chunks);
}